// OTG_CrossAttentionFusion_68813966016554
// MI455X (gfx1250) — compile-verified
//
#include <hip/hip_runtime.h>
#include <math.h>

#define BB 8
#define CC 64
#define NN 1024
#define NPP 1025
#define TC 192
#define A_I  (0.8f/1024.0f)
#define A_V  0.2f
#define EPSF 1e-6f

typedef float v2f __attribute__((ext_vector_type(2)));
typedef float v8f __attribute__((ext_vector_type(8)));

__device__ inline v8f wmma4(v2f a, v2f b, v8f c) {
  return __builtin_amdgcn_wmma_f32_16x16x4_f32(false, a, false, b, (short)0, c, false, false);
}
__device__ inline v8f vzero8() {
  v8f v;
#pragma unroll
  for (int i = 0; i < 8; ++i) v[i] = 0.0f;
  return v;
}

// ---------------- setup / small kernels ----------------

__global__ void init_k(unsigned* qstateU, float* qstateF, unsigned* hist) {
  int i = threadIdx.x;
  if (i == 0) {
    qstateU[0] = 0x7f7fffffu;      // running min bits (FLT_MAX)
    qstateU[1] = 0u;               // running max bits (0; cost >= 0)
    qstateF[4] = 786431.25f;       // target rank: 0.75*(N*N-1)
  }
  for (int j = i; j < 1024; j += blockDim.x) hist[j] = 0u;
}

__global__ void init_uv_k(float* uv, int n) {
  int i = blockIdx.x * blockDim.x + threadIdx.x;
  if (i < n) uv[i] = 1.0f;
}

__global__ void sq_k(const float* __restrict__ x, float* __restrict__ sq) {
  int idx = blockIdx.x * blockDim.x + threadIdx.x;   // B*N
  int n = idx & (NN - 1);
  int b = idx >> 10;
  const float* xb = x + ((size_t)b * CC) * NN + n;
  float acc = 0.0f;
#pragma unroll 8
  for (int c = 0; c < CC; ++c) { float v = xb[(size_t)c * NN]; acc += v * v; }
  sq[idx] = acc;
}

// ---------------- cost + K (WMMA f32 GEMM) ----------------
// block tile: 64 rows(i, src a) x 32 cols(j, src b); 8 waves = 4x2 tiles of 16x16
__global__ __launch_bounds__(256) void cost_k(
    const float* __restrict__ xa, const float* __restrict__ xb,
    const float* __restrict__ sqa, const float* __restrict__ sqb,
    float* __restrict__ Km, float* __restrict__ cost0) {
  const int lane = threadIdx.x & 31;
  const int wave = threadIdx.x >> 5;
  const int half = lane >> 4;
  const int l16 = lane & 15;
  const int jt = blockIdx.x % (NN / 32);
  const int it = (blockIdx.x / (NN / 32)) % (NN / 64);
  const int b  = blockIdx.x / ((NN / 32) * (NN / 64));
  const int i0 = it * 64 + (wave >> 1) * 16;
  const int j0 = jt * 32 + (wave & 1) * 16;
  const float* xab = xa + (size_t)b * CC * NN;
  const float* xbb = xb + (size_t)b * CC * NN;

  v2f af[16];
  const int arow = i0 + l16;
#pragma unroll
  for (int kk = 0; kk < 16; ++kk) {
    int ch0 = 4 * kk + 2 * half;
    af[kk].x = xab[(size_t)ch0 * NN + arow];
    af[kk].y = xab[(size_t)(ch0 + 1) * NN + arow];
  }
  v8f s = vzero8();
  const int bcol = j0 + l16;
#pragma unroll
  for (int kk = 0; kk < 16; ++kk) {
    int ch0 = 4 * kk + 2 * half;
    v2f bf;
    bf.x = xbb[(size_t)ch0 * NN + bcol];
    bf.y = xbb[(size_t)(ch0 + 1) * NN + bcol];
    s = wmma4(af[kk], bf, s);
  }
#pragma unroll
  for (int r = 0; r < 8; ++r) {
    int i = i0 + r + 8 * half;
    int j = j0 + l16;
    float c = sqa[(size_t)b * NN + i] + sqb[(size_t)b * NN + j] - 2.0f * s[r];
    c = fmaxf(c, 0.0f);
    Km[((size_t)b * NN + i) * NN + j] = fmaxf(__expf(-10.0f * c), 1e-8f);
    if (b == 0) cost0[(size_t)i * NN + j] = c;
  }
}

// ---------------- quantile (histogram refinement) ----------------

__global__ void minmax_k(const float* __restrict__ cost0, unsigned* qstateU) {
  float lmin = 3.4e38f, lmax = 0.0f;
  for (size_t i = blockIdx.x * blockDim.x + threadIdx.x; i < (size_t)NN * NN;
       i += (size_t)gridDim.x * blockDim.x) {
    float v = cost0[i];
    lmin = fminf(lmin, v);
    lmax = fmaxf(lmax, v);
  }
  for (int off = 16; off; off >>= 1) {
    lmin = fminf(lmin, __shfl_xor(lmin, off, 32));
    lmax = fmaxf(lmax, __shfl_xor(lmax, off, 32));
  }
  if ((threadIdx.x & 31) == 0) {
    atomicMin(&qstateU[0], __float_as_uint(lmin));   // nonneg floats: uint order == float order
    atomicMax(&qstateU[1], __float_as_uint(lmax));
  }
}

__global__ void seed_k(float* qstate) {
  unsigned* qu = (unsigned*)qstate;
  float mn = __uint_as_float(qu[0]);
  float mx = __uint_as_float(qu[1]);
  qstate[2] = mn;
  qstate[3] = mx + fmaxf(mx * 1e-6f, 1e-20f);
}

__global__ void hist_k(const float* __restrict__ cost0, const float* __restrict__ qstate,
                       unsigned* __restrict__ hist) {
  __shared__ unsigned hloc[1024];
  for (int i = threadIdx.x; i < 1024; i += blockDim.x) hloc[i] = 0u;
  __syncthreads();
  float lo = qstate[2], hi = qstate[3];
  float scale = 1024.0f / fmaxf(hi - lo, 1e-30f);
  for (size_t i = blockIdx.x * blockDim.x + threadIdx.x; i < (size_t)NN * NN;
       i += (size_t)gridDim.x * blockDim.x) {
    float v = cost0[i];
    if (v >= lo && v < hi) {
      int bi = (int)((v - lo) * scale);
      if (bi > 1023) bi = 1023;
      if (bi < 0) bi = 0;
      atomicAdd(&hloc[bi], 1u);
    }
  }
  __syncthreads();
  for (int i = threadIdx.x; i < 1024; i += blockDim.x)
    if (hloc[i]) atomicAdd(&hist[i], hloc[i]);
}

__global__ void select_k(float* qstate, unsigned* hist, int finalPass) {
  float lo = qstate[2], hi = qstate[3], t = qstate[4];
  float w = (hi - lo) * (1.0f / 1024.0f);
  unsigned target = (unsigned)t;
  unsigned cum = 0, cnt = 0;
  int k = 1023;
  for (int i = 0; i < 1024; ++i) {
    unsigned c = hist[i];
    if (cum + c > target) { k = i; cnt = c; break; }
    cum += c;
  }
  if (cnt == 0) cnt = 1;
  float nlo = lo + (float)k * w;
  float nt = t - (float)cum;
  if (finalPass) {
    float frac = (nt + 0.5f) / (float)cnt;
    frac = fminf(fmaxf(frac, 0.0f), 1.0f);
    float vq = nlo + w * frac;
    qstate[5] = vq;
    qstate[6] = fmaxf(__expf(-10.0f * vq), 1e-8f);   // kv
  } else {
    qstate[2] = nlo;
    qstate[3] = nlo + w;
    qstate[4] = nt;
  }
  for (int i = 0; i < 1024; ++i) hist[i] = 0u;
}

// ---------------- Sinkhorn matvecs (K is L2-resident) ----------------

__global__ __launch_bounds__(256) void row_mv_k(const float* __restrict__ Km,
                                                const float* __restrict__ x,
                                                float* __restrict__ out) {
  const int wave = threadIdx.x >> 5, lane = threadIdx.x & 31;
  const int row = blockIdx.x * 8 + wave;              // row = b*N + i
  const int b = row >> 10;
  const float* Kr = Km + (size_t)row * NN;
  const float* xb = x + (size_t)b * NPP;
  float acc = 0.0f;
#pragma unroll 4
  for (int j = lane; j < NN; j += 32) acc += Kr[j] * xb[j];
  for (int off = 16; off; off >>= 1) acc += __shfl_xor(acc, off, 32);
  if (lane == 0) out[row] = acc;
}

__global__ __launch_bounds__(256) void col_mv_k(const float* __restrict__ Km,
                                                const float* __restrict__ x,
                                                float* __restrict__ out) {
  __shared__ float red[256];
  const int tid = threadIdx.x, c = tid & 31, g = tid >> 5;
  const int b = blockIdx.x / (NN / 32);
  const int j0 = (blockIdx.x % (NN / 32)) * 32;
  const float* Kb = Km + (size_t)b * NN * NN;
  const float* xb = x + (size_t)b * NPP;
  float acc = 0.0f;
#pragma unroll 4
  for (int i = g; i < NN; i += 8) acc += Kb[(size_t)i * NN + j0 + c] * xb[i];
  red[tid] = acc;
  __syncthreads();
  if (g == 0) {
    float s = acc;
    for (int gg = 1; gg < 8; ++gg) s += red[gg * 32 + c];
    out[(size_t)b * NN + j0 + c] = s;
  }
}

__global__ __launch_bounds__(1024) void update_u_k(
    const float* __restrict__ ra, const float* __restrict__ cb,
    const float* __restrict__ v_a, const float* __restrict__ v_b,
    float* __restrict__ u_a, float* __restrict__ u_b,
    const float* __restrict__ qstate) {
  __shared__ float s1[32], s2[32];
  const int b = blockIdx.x, i = threadIdx.x;
  const float kv = qstate[6];
  const float vaN = v_a[(size_t)b * NPP + NN], vbN = v_b[(size_t)b * NPP + NN];
  float va = v_a[(size_t)b * NPP + i], vb = v_b[(size_t)b * NPP + i];
  u_a[(size_t)b * NPP + i] = A_I / (ra[(size_t)b * NN + i] + kv * vaN + EPSF);
  u_b[(size_t)b * NPP + i] = A_I / (cb[(size_t)b * NN + i] + kv * vbN + EPSF);
  float t1 = va, t2 = vb;
  for (int off = 16; off; off >>= 1) { t1 += __shfl_xor(t1, off, 32); t2 += __shfl_xor(t2, off, 32); }
  if ((i & 31) == 0) { s1[i >> 5] = t1; s2[i >> 5] = t2; }
  __syncthreads();
  if (i == 0) {
    float S1 = 0.0f, S2 = 0.0f;
    for (int w = 0; w < 32; ++w) { S1 += s1[w]; S2 += s2[w]; }
    u_a[(size_t)b * NPP + NN] = A_V / (kv * S1 + vaN + EPSF);
    u_b[(size_t)b * NPP + NN] = A_V / (kv * S2 + vbN + EPSF);
  }
}

__global__ __launch_bounds__(1024) void update_v_k(
    const float* __restrict__ ca, const float* __restrict__ rb,
    const float* __restrict__ u_a, const float* __restrict__ u_b,
    float* __restrict__ v_a, float* __restrict__ v_b,
    const float* __restrict__ qstate) {
  __shared__ float s1[32], s2[32];
  const int b = blockIdx.x, i = threadIdx.x;
  const float kv = qstate[6];
  const float uaN = u_a[(size_t)b * NPP + NN], ubN = u_b[(size_t)b * NPP + NN];
  float ua = u_a[(size_t)b * NPP + i], ub = u_b[(size_t)b * NPP + i];
  v_a[(size_t)b * NPP + i] = A_I / (ca[(size_t)b * NN + i] + kv * uaN + EPSF);
  v_b[(size_t)b * NPP + i] = A_I / (rb[(size_t)b * NN + i] + kv * ubN + EPSF);
  float t1 = ua, t2 = ub;
  for (int off = 16; off; off >>= 1) { t1 += __shfl_xor(t1, off, 32); t2 += __shfl_xor(t2, off, 32); }
  if ((i & 31) == 0) { s1[i >> 5] = t1; s2[i >> 5] = t2; }
  __syncthreads();
  if (i == 0) {
    float S1 = 0.0f, S2 = 0.0f;
    for (int w = 0; w < 32; ++w) { S1 += s1[w]; S2 += s2[w]; }
    v_a[(size_t)b * NPP + NN] = A_V / (kv * S1 + uaN + EPSF);
    v_b[(size_t)b * NPP + NN] = A_V / (kv * S2 + ubN + EPSF);
  }
}

// ---------------- QKV ----------------

__global__ void pw_k(const float* __restrict__ x, const float* __restrict__ w,
                     float* __restrict__ y) {
  int idx = blockIdx.x * blockDim.x + threadIdx.x;     // B*192*N
  int n = idx & (NN - 1);
  int o = (idx >> 10) % TC;
  int b = idx / (TC * NN);
  const float* xb = x + (size_t)b * CC * NN + n;
  const float* wr = w + (size_t)o * CC;
  float acc = 0.0f;
#pragma unroll 8
  for (int c = 0; c < CC; ++c) acc += wr[c] * xb[(size_t)c * NN];
  y[idx] = acc;
}

__global__ void dw_k(const float* __restrict__ y, const float* __restrict__ dw,
                     float* __restrict__ qk, float* __restrict__ vT) {
  int idx = blockIdx.x * blockDim.x + threadIdx.x;
  int n = idx & (NN - 1);
  int o = (idx >> 10) % TC;
  int b = idx / (TC * NN);
  int h = n >> 5, wI = n & 31;
  const float* src = y + ((size_t)b * TC + o) * NN;
  const float* wt = dw + (size_t)o * 9;
  float acc = 0.0f;
#pragma unroll
  for (int dy = -1; dy <= 1; ++dy)
#pragma unroll
    for (int dx = -1; dx <= 1; ++dx) {
      int hh = h + dy, ww = wI + dx;
      if (hh >= 0 && hh < 32 && ww >= 0 && ww < 32)
        acc += wt[(dy + 1) * 3 + (dx + 1)] * src[hh * 32 + ww];
    }
  if (o < 2 * CC) qk[((size_t)b * 2 * CC + o) * NN + n] = acc;        // q: 0..63, k: 64..127
  else            vT[((size_t)b * NN + n) * CC + (o - 2 * CC)] = acc; // V transposed [B,N,C]
}

// ---------------- flash cross-attention with OT bias (WMMA f32) ----------------
// block = 8 waves, each wave owns a 16-row query tile -> 128 rows/block.
__global__ __launch_bounds__(256) void attn_k(
    const float* __restrict__ qbuf,   // [B,128,N], q in ch 0..63
    const float* __restrict__ kbuf,   // [B,128,N], k in ch 64..127
    const float* __restrict__ vT,     // [B,N,64]
    const float* __restrict__ Km,     // [B,N,N]
    const float* __restrict__ u,      // [B,NPP]
    const float* __restrict__ S,      // [B,N]
    const float* __restrict__ coefp,
    int transposeK,
    float* __restrict__ ctx) {        // [B,N,64]
  __shared__ float pshare[8 * 256];
  const int lane = threadIdx.x & 31;
  const int wave = threadIdx.x >> 5;
  const int half = lane >> 4;
  const int l16 = lane & 15;
  const int b = blockIdx.x / (NN / 128);
  const int i0 = (blockIdx.x % (NN / 128)) * 128 + wave * 16;
  const float coef = *coefp;

  const float* qb = qbuf + (size_t)b * 2 * CC * NN;
  const float* kb = kbuf + (size_t)b * 2 * CC * NN + (size_t)CC * NN;
  const float* vb = vT + (size_t)b * NN * CC;
  const float* Kb = Km + (size_t)b * NN * NN;
  const float* ub = u + (size_t)b * NPP;
  float* psl = pshare + wave * 256;

  // Q fragments (16 k-steps over C=64)
  v2f qf[16];
  const int qrow = i0 + l16;
#pragma unroll
  for (int kk = 0; kk < 16; ++kk) {
    int ch0 = 4 * kk + 2 * half;
    qf[kk].x = qb[(size_t)ch0 * NN + qrow];
    qf[kk].y = qb[(size_t)(ch0 + 1) * NN + qrow];
  }
  float Sinv[8], mrow[8], lrow[8];
#pragma unroll
  for (int r = 0; r < 8; ++r) {
    Sinv[r] = 1.0f / S[(size_t)b * NN + i0 + r + 8 * half];
    mrow[r] = -3.0e38f;
    lrow[r] = 0.0f;
  }
  v8f acc[4];
#pragma unroll
  for (int t = 0; t < 4; ++t) acc[t] = vzero8();

  for (int jt = 0; jt < NN / 16; ++jt) {
    const int j0 = jt * 16;
    // S = Q * K^T (16x16 tile)
    v8f sc = vzero8();
    const int kcol = j0 + l16;
#pragma unroll
    for (int kk = 0; kk < 16; ++kk) {
      int ch0 = 4 * kk + 2 * half;
      v2f bf;
      bf.x = kb[(size_t)ch0 * NN + kcol];
      bf.y = kb[(size_t)(ch0 + 1) * NN + kcol];
      sc = wmma4(qf[kk], bf, sc);
    }
    const float uj = ub[j0 + l16];
    float p[8];
#pragma unroll
    for (int r = 0; r < 8; ++r) {
      int i = i0 + r + 8 * half;
      int j = j0 + l16;
      float kval = transposeK ? Kb[(size_t)j * NN + i] : Kb[(size_t)i * NN + j];
      float logit = sc[r] * 0.125f + coef * kval * uj * Sinv[r];
      float tm = logit;
      for (int off = 1; off < 16; off <<= 1) tm = fmaxf(tm, __shfl_xor(tm, off, 32));
      float mnew = fmaxf(mrow[r], tm);
      float scl = __expf(mrow[r] - mnew);
      float pe = __expf(logit - mnew);
      float ps = pe;
      for (int off = 1; off < 16; off <<= 1) ps += __shfl_xor(ps, off, 32);
      lrow[r] = lrow[r] * scl + ps;
      mrow[r] = mnew;
      p[r] = pe;
#pragma unroll
      for (int t = 0; t < 4; ++t) acc[t][r] *= scl;
    }
    // stage P in wave-private LDS to re-lay out D->A fragments
#pragma unroll
    for (int r = 0; r < 8; ++r) psl[(r + 8 * half) * 16 + l16] = p[r];
    asm volatile("" ::: "memory");
    // ctx += P * V   (K-dim = 16 j's, 4 steps; 4 c-tiles)
#pragma unroll
    for (int kk2 = 0; kk2 < 4; ++kk2) {
      int jj0 = 4 * kk2 + 2 * half;
      v2f pa;
      pa.x = psl[l16 * 16 + jj0];
      pa.y = psl[l16 * 16 + jj0 + 1];
      const float* vr0 = vb + (size_t)(j0 + jj0) * CC;
      const float* vr1 = vb + (size_t)(j0 + jj0 + 1) * CC;
#pragma unroll
      for (int t = 0; t < 4; ++t) {
        v2f bv;
        bv.x = vr0[t * 16 + l16];
        bv.y = vr1[t * 16 + l16];
        acc[t] = wmma4(pa, bv, acc[t]);
      }
    }
    asm volatile("" ::: "memory");
  }
#pragma unroll
  for (int r = 0; r < 8; ++r) {
    int i = i0 + r + 8 * half;
    float linv = 1.0f / lrow[r];
#pragma unroll
    for (int t = 0; t < 4; ++t)
      ctx[((size_t)b * NN + i) * CC + t * 16 + l16] = acc[t][r] * linv;
  }
}

// ---------------- output projection + bias + residual ----------------

__global__ __launch_bounds__(256) void proj_k(
    const float* __restrict__ ctx, const float* __restrict__ W,
    const float* __restrict__ bias, const float* __restrict__ identity,
    float* __restrict__ out) {
  __shared__ float lctx[4 * CC];
  const int o = threadIdx.x & 63, nn = threadIdx.x >> 6;
  const int b = blockIdx.x / (NN / 4);
  const int n0 = (blockIdx.x % (NN / 4)) * 4;
  lctx[nn * CC + o] = ctx[((size_t)b * NN + n0 + nn) * CC + o];
  __syncthreads();
  float acc = bias[o];
  const float* wr = W + (size_t)o * CC;
#pragma unroll 8
  for (int c = 0; c < CC; ++c) acc += wr[c] * lctx[nn * CC + c];
  const int n = n0 + nn;
  out[((size_t)b * CC + o) * NN + n] = acc + identity[((size_t)b * CC + o) * NN + n];
}

// ---------------- host launch ----------------

extern "C" void kernel_launch(void* const* d_in, const int* in_sizes, int n_in,
                              void* d_out, int out_size, void* d_ws, size_t ws_size,
                              hipStream_t stream) {
  (void)in_sizes; (void)n_in; (void)out_size; (void)ws_size;
  const float* inp_a = (const float*)d_in[0];
  const float* inp_b = (const float*)d_in[1];
  const float* alpha = (const float*)d_in[2];
  const float* beta  = (const float*)d_in[3];
  const float* w1_A  = (const float*)d_in[4];
  const float* dw_A  = (const float*)d_in[5];
  const float* outA_w = (const float*)d_in[6];
  const float* outA_b = (const float*)d_in[7];
  const float* w1_B  = (const float*)d_in[8];
  const float* dw_B  = (const float*)d_in[9];
  const float* outB_w = (const float*)d_in[10];
  const float* outB_b = (const float*)d_in[11];
  float* out = (float*)d_out;

  float* f = (float*)d_ws;
  size_t o = 0;
  float* qstate = f + o; o += 16;
  unsigned* hist = (unsigned*)(f + o); o += 1024;
  float* u_a = f + o; o += (size_t)BB * NPP;
  float* v_a = f + o; o += (size_t)BB * NPP;
  float* u_b = f + o; o += (size_t)BB * NPP;
  float* v_b = f + o; o += (size_t)BB * NPP;
  float* ra = f + o; o += (size_t)BB * NN;
  float* rb = f + o; o += (size_t)BB * NN;
  float* ca = f + o; o += (size_t)BB * NN;
  float* cb = f + o; o += (size_t)BB * NN;
  float* sqa = f + o; o += (size_t)BB * NN;
  float* sqb = f + o; o += (size_t)BB * NN;
  float* cost0 = f + o; o += (size_t)NN * NN;
  float* Km = f + o; o += (size_t)BB * NN * NN;
  float* pw = f + o; o += (size_t)BB * TC * NN;
  float* qkA = f + o; o += (size_t)BB * 2 * CC * NN;
  float* qkB = f + o; o += (size_t)BB * 2 * CC * NN;
  float* vAT = f + o; o += (size_t)BB * NN * CC;
  float* vBT = f + o; o += (size_t)BB * NN * CC;
  float* ctxA = f + o; o += (size_t)BB * NN * CC;
  float* ctxB = f + o; o += (size_t)BB * NN * CC;

  // setup
  init_k<<<1, 256, 0, stream>>>((unsigned*)qstate, qstate, hist);
  {
    int nuv = 4 * BB * NPP;
    init_uv_k<<<(nuv + 255) / 256, 256, 0, stream>>>(u_a, nuv); // u_a..v_b contiguous
  }
  sq_k<<<BB * NN / 256, 256, 0, stream>>>(inp_a, sqa);
  sq_k<<<BB * NN / 256, 256, 0, stream>>>(inp_b, sqb);

  // cost + K (WMMA)
  cost_k<<<BB * (NN / 64) * (NN / 32), 256, 0, stream>>>(inp_a, inp_b, sqa, sqb, Km, cost0);

  // quantile of cost0 -> vq, kv
  minmax_k<<<512, 256, 0, stream>>>(cost0, (unsigned*)qstate);
  seed_k<<<1, 1, 0, stream>>>(qstate);
  for (int p = 0; p < 3; ++p) {
    hist_k<<<512, 256, 0, stream>>>(cost0, qstate, hist);
    select_k<<<1, 1, 0, stream>>>(qstate, hist, (p == 2) ? 1 : 0);
  }

  // Sinkhorn: 50 iterations; plan-a uses K rows, plan-b uses K columns (K_b = K^T)
  for (int it = 0; it < 50; ++it) {
    row_mv_k<<<BB * NN / 8, 256, 0, stream>>>(Km, v_a, ra);   // K  v_a
    col_mv_k<<<BB * NN / 32, 256, 0, stream>>>(Km, v_b, cb);  // K^T v_b
    update_u_k<<<BB, 1024, 0, stream>>>(ra, cb, v_a, v_b, u_a, u_b, qstate);
    row_mv_k<<<BB * NN / 8, 256, 0, stream>>>(Km, u_b, rb);   // K  u_b  (-> S_b after last iter)
    col_mv_k<<<BB * NN / 32, 256, 0, stream>>>(Km, u_a, ca);  // K^T u_a (-> S_a after last iter)
    update_v_k<<<BB, 1024, 0, stream>>>(ca, rb, u_a, u_b, v_a, v_b, qstate);
  }

  // QKV for both nets (pw buffer reused sequentially)
  pw_k<<<BB * TC * NN / 256, 256, 0, stream>>>(inp_a, w1_A, pw);
  dw_k<<<BB * TC * NN / 256, 256, 0, stream>>>(pw, dw_A, qkA, vAT);
  pw_k<<<BB * TC * NN / 256, 256, 0, stream>>>(inp_b, w1_B, pw);
  dw_k<<<BB * TC * NN / 256, 256, 0, stream>>>(pw, dw_B, qkB, vBT);

  // out_a = attn(qB, kA, vA) + alpha * trans_a ; trans_a[i,j] = K[j,i]*u_a[j]/S_a[i]
  attn_k<<<BB * (NN / 128), 256, 0, stream>>>(qkB, qkA, vAT, Km, u_a, ca, alpha, 1, ctxA);
  // out_b = attn(qA, kB, vB) + beta  * trans_b ; trans_b[i,j] = K[i,j]*u_b[j]/S_b[i]
  attn_k<<<BB * (NN / 128), 256, 0, stream>>>(qkA, qkB, vBT, Km, u_b, rb, beta, 0, ctxB);

  // projection + bias + residual
  proj_k<<<BB * NN / 4, 256, 0, stream>>>(ctxA, outA_w, outA_b, inp_a, out);
  proj_k<<<BB * NN / 4, 256, 0, stream>>>(ctxB, outB_w, outB_b, inp_b, out + (size_t)BB * CC * NN);
}